// CRFPerTaskGroupOutputLayer_18236431139096
// MI455X (gfx1250) — compile-verified
//
#include <hip/hip_runtime.h>
#include <cstddef>
#include <cstdint>

// ---------------------------------------------------------------------------
// Problem constants (from reference): B=64, T=512, D=1024, L=16
// ---------------------------------------------------------------------------
#define BD 64
#define TD 512
#define DD 1024
#define LD 16
#define MTOT (BD * TD)          // 32768 rows of the logits GEMM
#define MTILES (MTOT / 16)      // 2048 16-row tiles
#define WAVES_PER_BLOCK 8       // 256 threads = 8 wave32

typedef float v2f __attribute__((ext_vector_type(2)));
typedef float v8f __attribute__((ext_vector_type(8)));
typedef unsigned int v4u __attribute__((ext_vector_type(4)));
typedef int v4i __attribute__((ext_vector_type(4)));
typedef int v8i __attribute__((ext_vector_type(8)));

// ---------------------------------------------------------------------------
// Kernel 1: logits[b,t,l] = x[b,t,:] @ W[:,l] + b[l]
// One wave32 per 16-row M-tile, K-loop of V_WMMA_F32_16X16X4_F32.
// W (1024x16 f32 = 64KB, contiguous) staged into LDS once per block by the
// Tensor Data Mover (TENSOR_LOAD_TO_LDS + s_wait_tensorcnt), issued by wave 0
// only; TDM ignores EXEC and frees the vector pipes during staging.
//
// A-operand (16x4 f32) lane layout: lanes 0-15 / 16-31 hold M=lane&15,
//   VGPR0/1 = K = 2*(lane>>4) + {0,1}  -> one float2 global load per step.
// B-operand (4x16 f32): VGPR0/1 = row K = 2*(lane>>4)+{0,1}, col N=lane&15
//   -> ds_load_2addr_b32 per step, conflict-free (consecutive dwords per row).
// C/D (16x16 f32, 8 VGPRs): row = v + 8*(lane>>4), col = lane&15.
// ---------------------------------------------------------------------------
__global__ __launch_bounds__(256) void gemm_logits_kernel(
    const float* __restrict__ x, const float* __restrict__ W,
    const float* __restrict__ bias, float* __restrict__ logits) {
  __shared__ float Ws[DD * LD];  // 64 KB

  const int tid = threadIdx.x;

#if __has_builtin(__builtin_amdgcn_tensor_load_to_lds) && \
    __has_builtin(__builtin_amdgcn_s_wait_tensorcnt)
  if (tid < 32) {
    // --- Tensor DMA descriptor (D#) for a 1-row x 16384-elem f32 tile ---
    const uint64_t ga = (uint64_t)(uintptr_t)W;
    // Generic LDS pointers are {shared_aperture, offset}; low 32 bits are the
    // LDS byte offset the D# wants.
    const uint32_t lds_off = (uint32_t)(uintptr_t)&Ws[0];

    v4u g0;
    g0[0] = 1u;                  // count=1 valid descriptor; flags 0
    g0[1] = lds_off;             // lds_addr
    g0[2] = (uint32_t)ga;        // global_addr[31:0]
    g0[3] = (uint32_t)((ga >> 32) & 0x1FFFFFFu) | (2u << 30);  // addr[56:32]|type=2

    v8i g1;
    g1[0] = (int)(2u << 16);               // workgroup_mask=0, data_size=2 (4B)
    g1[1] = (int)((DD * LD & 0xFFFFu) << 16);  // tensor_dim0[15:0] in bits 63:48
    g1[2] = (int)(1u << 16);               // tensor_dim0[31:16]=0, tensor_dim1=1
    g1[3] = (int)((DD * LD & 0xFFFFu) << 16);  // tile_dim0 = 16384 (bits 127:112)
    g1[4] = 0;                             // tile_dim1=0 (unused), tile_dim2=0
    g1[5] = (int)(DD * LD);                // tensor_dim0_stride low
    g1[6] = 0;
    g1[7] = 0;

    v4i gz = {0, 0, 0, 0};  // groups 2/3 unused (<=2D tensor)

#if defined(__clang_major__) && __clang_major__ >= 23
    v8i gz8 = {0, 0, 0, 0, 0, 0, 0, 0};
    __builtin_amdgcn_tensor_load_to_lds(g0, g1, gz, gz, gz8, 0);
#else
    __builtin_amdgcn_tensor_load_to_lds(g0, g1, gz, gz, 0);
#endif
    __builtin_amdgcn_s_wait_tensorcnt(0);
  }
#else
  {
    const float4* __restrict__ Wg = (const float4*)W;
    float4* Wl = (float4*)Ws;
#pragma unroll
    for (int i = 0; i < (DD * LD / 4) / 256; ++i)
      Wl[tid + i * 256] = Wg[tid + i * 256];
  }
#endif
  __syncthreads();

  const int wave = tid >> 5;
  const int lane = tid & 31;
  const int n = lane & 15;   // B/C/D column; also A row-within-tile
  const int kh = lane >> 4;  // lane-half: selects K pair {0,1} vs {2,3}

  const int tile = blockIdx.x * WAVES_PER_BLOCK + wave;  // 0..2047
  const size_t row = (size_t)tile * 16 + (size_t)n;
  const float* __restrict__ xrow = x + row * DD;

  const float bn = bias[n];
  v8f c = {};

  for (int kc = 0; kc < DD; kc += 256) {
    if (kc + 256 < DD) {
      // streaming hint for the next x chunk (global_prefetch_b8)
      __builtin_prefetch(xrow + kc + 256, 0, 1);
    }
#pragma unroll 4
    for (int k0 = kc; k0 < kc + 256; k0 += 4) {
      const float2 av = *(const float2*)(xrow + k0 + 2 * kh);
      v2f a;
      a.x = av.x;
      a.y = av.y;
      v2f bb;
      bb.x = Ws[(k0 + 2 * kh + 0) * LD + n];
      bb.y = Ws[(k0 + 2 * kh + 1) * LD + n];
      // D = A*B + C, f32 in / f32 out (exact -- matches f32 reference)
      c = __builtin_amdgcn_wmma_f32_16x16x4_f32(
          /*neg_a=*/false, a, /*neg_b=*/false, bb,
          /*c_mod=*/(short)0, c, /*reuse_a=*/false, /*reuse_b=*/false);
    }
  }

  float* __restrict__ out = logits + (size_t)tile * 16 * LD;
#pragma unroll
  for (int v = 0; v < 8; ++v) {
    const int m = v + 8 * kh;
    out[m * LD + n] = c[v] + bn;
  }
}

// ---------------------------------------------------------------------------
// Kernel 2: CRF forward scan + path score, one wave32 per batch row.
// Lane j = lane&15 owns alpha[j]; the inner logsumexp over i is split across
// lane-halves (h = lane>>4 handles i in [8h, 8h+8)) to halve the serial
// exp-chain of the 511-step dependent scan; halves merge via shfl_xor(16).
// ---------------------------------------------------------------------------
__global__ __launch_bounds__(32) void crf_scan_kernel(
    const float* __restrict__ logits, const unsigned char* __restrict__ mask,
    const int* __restrict__ labels, const float* __restrict__ trans,
    const float* __restrict__ start_trans, const float* __restrict__ end_trans,
    float* __restrict__ partial) {
  const int b = blockIdx.x;
  const int lane = threadIdx.x;
  const int j = lane & 15;
  const int h = lane >> 4;
  const int base = h * 8;

  // This lane's slice of the transition matrix column j: trans[base+ii][j]
  float tc[8];
#pragma unroll
  for (int ii = 0; ii < 8; ++ii) tc[ii] = trans[(base + ii) * LD + j];

  const float* __restrict__ lg = logits + (size_t)b * TD * LD;
  const unsigned char* __restrict__ mk = mask + (size_t)b * TD;
  const int* __restrict__ lb = labels + (size_t)b * TD;

  float alpha = start_trans[j] + lg[j];  // duplicated across both halves

  for (int t = 1; t < TD; ++t) {
    const float emit = lg[t * LD + j];
    float v[8];
#pragma unroll
    for (int ii = 0; ii < 8; ++ii)
      v[ii] = __shfl(alpha, base + ii, 32) + tc[ii];
    float m = v[0];
#pragma unroll
    for (int ii = 1; ii < 8; ++ii) m = fmaxf(m, v[ii]);
    float s = 0.0f;
#pragma unroll
    for (int ii = 0; ii < 8; ++ii) s += __expf(v[ii] - m);
    // merge the two i-halves (lane j <-> lane j+16)
    const float mo = __shfl_xor(m, 16, 32);
    const float so = __shfl_xor(s, 16, 32);
    const float M = fmaxf(m, mo);
    s = s * __expf(m - M) + so * __expf(mo - M);
    const float na = M + __logf(s) + emit;
    alpha = mk[t] ? na : alpha;
  }

  // log_z = logsumexp_j(alpha[j] + end_trans[j]); butterfly within 16-group
  const float xz = alpha + end_trans[j];
  float mz = xz;
#pragma unroll
  for (int off = 8; off >= 1; off >>= 1)
    mz = fmaxf(mz, __shfl_xor(mz, off, 32));
  float sz = __expf(xz - mz);
#pragma unroll
  for (int off = 8; off >= 1; off >>= 1) sz += __shfl_xor(sz, off, 32);
  const float log_z = mz + __logf(sz);

  // Gold-path score: strided over t across all 32 lanes.
  float es = 0.0f, ts = 0.0f;
  int cnt = 0;
  for (int t = lane; t < TD; t += 32) {
    const int lab = lb[t];
    if (mk[t]) {
      es += lg[t * LD + lab];
      cnt += 1;
      if (t > 0) ts += trans[lb[t - 1] * LD + lab];
    }
  }
#pragma unroll
  for (int off = 16; off >= 1; off >>= 1) {
    es += __shfl_xor(es, off, 32);
    ts += __shfl_xor(ts, off, 32);
    cnt += __shfl_xor(cnt, off, 32);
  }

  if (lane == 0) {
    const int last_idx = cnt - 1;
    const int last_tag = lb[last_idx];
    const float score = start_trans[lb[0]] + es + ts + end_trans[last_tag];
    partial[b] = score - log_z;
  }
}

// ---------------------------------------------------------------------------
// Kernel 3: deterministic final reduction: out = -sum_b(score_b - log_z_b)
// (single-thread ordered sum -> bit-stable across graph replays)
// ---------------------------------------------------------------------------
__global__ void final_reduce_kernel(const float* __restrict__ partial,
                                    float* __restrict__ out) {
  if (threadIdx.x == 0 && blockIdx.x == 0) {
    float s = 0.0f;
    for (int i = 0; i < BD; ++i) s += partial[i];
    out[0] = -s;
  }
}

// ---------------------------------------------------------------------------
// Launch: inputs in setup_inputs() order:
//   0:x(f32) 1:mask(bool/u8) 2:labels(i32) 3:W(f32) 4:b(f32)
//   5:transitions(f32) 6:start_trans(f32) 7:end_trans(f32)
// Workspace: [logits: 32768*16 f32 = 2MB][partial: 64 f32]
// ---------------------------------------------------------------------------
extern "C" void kernel_launch(void* const* d_in, const int* in_sizes, int n_in,
                              void* d_out, int out_size, void* d_ws,
                              size_t ws_size, hipStream_t stream) {
  (void)in_sizes;
  (void)n_in;
  (void)out_size;
  (void)ws_size;

  const float* x = (const float*)d_in[0];
  const unsigned char* mask = (const unsigned char*)d_in[1];
  const int* labels = (const int*)d_in[2];
  const float* W = (const float*)d_in[3];
  const float* bias = (const float*)d_in[4];
  const float* trans = (const float*)d_in[5];
  const float* start_trans = (const float*)d_in[6];
  const float* end_trans = (const float*)d_in[7];

  float* logits = (float*)d_ws;
  float* partial = logits + (size_t)MTOT * LD;

  gemm_logits_kernel<<<MTILES / WAVES_PER_BLOCK, 256, 0, stream>>>(x, W, bias,
                                                                   logits);
  crf_scan_kernel<<<BD, 32, 0, stream>>>(logits, mask, labels, trans,
                                         start_trans, end_trans, partial);
  final_reduce_kernel<<<1, 1, 0, stream>>>(partial, (float*)d_out);
}